// LargeGNNRecommender_89481348645687
// MI455X (gfx1250) — compile-verified
//
#include <hip/hip_runtime.h>
#include <hip/hip_bf16.h>

typedef float v2f __attribute__((ext_vector_type(2)));
typedef float v8f __attribute__((ext_vector_type(8)));

#if __has_builtin(__builtin_amdgcn_wmma_f32_16x16x4_f32)
#define HAVE_WMMA_F32X4 1
#else
#define HAVE_WMMA_F32X4 0
#endif

// ---------------------------------------------------------------------------
// Kernel 1: embedding gather  h0[i,:] = emb_table[x[i],:]
// 32 lanes per node, one float4 per lane (128 floats per row).
// ---------------------------------------------------------------------------
__global__ void gnn_gather_embed(const int* __restrict__ x,
                                 const float* __restrict__ table,
                                 float* __restrict__ h, int n) {
  int t = blockIdx.x * blockDim.x + threadIdx.x;
  int node = t >> 5;
  if (node >= n) return;
  int lane = t & 31;
  const float4* srcp = (const float4*)(table + (size_t)x[node] * 128);
  float4* dstp = (float4*)(h + (size_t)node * 128);
  dstp[lane] = srcp[lane];
}

// ---------------------------------------------------------------------------
// Kernel 2: edge scatter-add.  One wave32 per edge; lane k moves floats
// [4k,4k+4) of the source row into agg[dst] via f32 atomics (no-return ->
// global_atomic_add_f32, STOREcnt path).  Lane 0 bumps the degree count.
// ---------------------------------------------------------------------------
__global__ void gnn_scatter_edges(const float* __restrict__ h,
                                  const int* __restrict__ src,
                                  const int* __restrict__ dst,
                                  float* __restrict__ agg,
                                  float* __restrict__ cnt, int E) {
  int t = blockIdx.x * blockDim.x + threadIdx.x;
  int e = t >> 5;
  if (e >= E) return;
  int lane = t & 31;
  int s = src[e];
  int d = dst[e];
  float4 v = ((const float4*)(h + (size_t)s * 128))[lane];
  float* o = agg + (size_t)d * 128 + lane * 4;
  atomicAdd(o + 0, v.x);
  atomicAdd(o + 1, v.y);
  atomicAdd(o + 2, v.z);
  atomicAdd(o + 3, v.w);
  if (lane == 0) atomicAdd(cnt + d, 1.0f);
}

// ---------------------------------------------------------------------------
// Kernel 3: fused SAGE linear:
//   out = relu( (agg/max(cnt,1)) @ Wl^T + b + h @ Wr^T )
// One wave per 16-row tile. K=128 via 32 x V_WMMA_F32_16X16X4_F32 per operand
// matrix, 8 column tiles of 16. f32 A/B fragments are contiguous float2 pairs:
//   A vgpr layout (16x4 f32): lanes 0-15 K={0,1}, lanes 16-31 K={2,3}
//   B vgpr layout (4x16 f32): lane l -> column l, same K split.
// Mean-normalization folded into the A fragment (row-constant scale).
// ---------------------------------------------------------------------------
__global__ __launch_bounds__(160) void gnn_sage_linear_relu(
    const float* __restrict__ agg, const float* __restrict__ cnt,
    const float* __restrict__ h, const float* __restrict__ Wl,
    const float* __restrict__ Wr, const float* __restrict__ bias,
    float* __restrict__ out, int nTiles, int n) {
  const int wave = (blockIdx.x * blockDim.x + threadIdx.x) >> 5;
  if (wave >= nTiles) return;  // wave-uniform guard (EXEC stays all-ones)
  const int lane = threadIdx.x & 31;
  const int l = lane & 15;
  const int half = lane >> 4;
  const int row0 = wave * 16;
  int rowA = row0 + l;
  if (rowA >= n) rowA = n - 1;  // clamp reads; stores guarded below

  const float inv = 1.0f / fmaxf(cnt[rowA], 1.0f);
  const float* aggRow = agg + (size_t)rowA * 128;
  const float* hRow = h + (size_t)rowA * 128;

  for (int ct = 0; ct < 8; ++ct) {
    const int col0 = ct * 16;
    const float bv = bias[col0 + l];
    v8f c;
#pragma unroll
    for (int p = 0; p < 8; ++p) c[p] = bv;

#if HAVE_WMMA_F32X4
    // ---- agg(normalized) @ Wl^T ----
#pragma unroll
    for (int k0 = 0; k0 < 128; k0 += 4) {
      const int kk = k0 + 2 * half;
      v2f a, b;
      a[0] = aggRow[kk] * inv;
      a[1] = aggRow[kk + 1] * inv;
      const float* w = Wl + (size_t)(col0 + l) * 128 + kk;
      b[0] = w[0];
      b[1] = w[1];
      c = __builtin_amdgcn_wmma_f32_16x16x4_f32(false, a, false, b, (short)0,
                                                c, false, false);
    }
    // ---- h @ Wr^T ----
#pragma unroll
    for (int k0 = 0; k0 < 128; k0 += 4) {
      const int kk = k0 + 2 * half;
      v2f a, b;
      a[0] = hRow[kk];
      a[1] = hRow[kk + 1];
      const float* w = Wr + (size_t)(col0 + l) * 128 + kk;
      b[0] = w[0];
      b[1] = w[1];
      c = __builtin_amdgcn_wmma_f32_16x16x4_f32(false, a, false, b, (short)0,
                                                c, false, false);
    }
#else
    // Scalar fallback (only if the f32 WMMA builtin is unavailable).
#pragma unroll 1
    for (int p = 0; p < 8; ++p) {
      const int r = row0 + p + 8 * half;
      const int rc = r < n ? r : n - 1;
      const float irow = 1.0f / fmaxf(cnt[rc], 1.0f);
      float accv = bv;
      for (int k = 0; k < 128; ++k) {
        accv += agg[(size_t)rc * 128 + k] * irow * Wl[(size_t)(col0 + l) * 128 + k];
        accv += h[(size_t)rc * 128 + k] * Wr[(size_t)(col0 + l) * 128 + k];
      }
      c[p] = accv;
    }
#endif

    // ReLU + store.  D layout: vgpr p -> row (row0 + p + 8*half), col (col0+l).
#pragma unroll
    for (int p = 0; p < 8; ++p) {
      const int r = row0 + p + 8 * half;
      if (r < n) out[(size_t)r * 128 + col0 + l] = fmaxf(c[p], 0.0f);
    }
  }
}

// ---------------------------------------------------------------------------
// Host-side orchestration (graph-capture safe: only async ops on `stream`).
// ---------------------------------------------------------------------------
extern "C" void kernel_launch(void* const* d_in, const int* in_sizes, int n_in,
                              void* d_out, int out_size, void* d_ws,
                              size_t ws_size, hipStream_t stream) {
  const int* x = (const int*)d_in[0];
  const int* ei = (const int*)d_in[1];
  const float* emb = (const float*)d_in[2];
  const float* W1l = (const float*)d_in[3];
  const float* b1 = (const float*)d_in[4];
  const float* W1r = (const float*)d_in[5];
  const float* W2l = (const float*)d_in[6];
  const float* b2 = (const float*)d_in[7];
  const float* W2r = (const float*)d_in[8];

  const int N = in_sizes[0];
  const int E = in_sizes[1] / 2;
  const int* src = ei;
  const int* dst = ei + E;

  float* h0 = (float*)d_ws;                  // [N,128]
  float* agg = h0 + (size_t)N * 128;         // [N,128]
  float* h1 = agg + (size_t)N * 128;         // [N,128]
  float* cnt = h1 + (size_t)N * 128;         // [N]
  float* out = (float*)d_out;

  const size_t featBytes = (size_t)N * 128 * sizeof(float);
  const int nTiles = (N + 15) / 16;           // 3125 for N=50000 (= 5^5)
  const int linBlocks = (nTiles + 4) / 5;     // 5 waves/block, blockDim=160
  const int gatherBlocks = (N * 32 + 255) / 256;
  const int scatterBlocks = (int)(((long long)E * 32 + 255) / 256);

  // ---- layer 1 ----
  hipMemsetAsync(agg, 0, featBytes, stream);
  hipMemsetAsync(cnt, 0, (size_t)N * sizeof(float), stream);
  gnn_gather_embed<<<gatherBlocks, 256, 0, stream>>>(x, emb, h0, N);
  gnn_scatter_edges<<<scatterBlocks, 256, 0, stream>>>(h0, src, dst, agg, cnt, E);
  gnn_sage_linear_relu<<<linBlocks, 160, 0, stream>>>(agg, cnt, h0, W1l, W1r,
                                                      b1, h1, nTiles, N);

  // ---- layer 2 ----
  hipMemsetAsync(agg, 0, featBytes, stream);
  hipMemsetAsync(cnt, 0, (size_t)N * sizeof(float), stream);
  gnn_scatter_edges<<<scatterBlocks, 256, 0, stream>>>(h1, src, dst, agg, cnt, E);
  gnn_sage_linear_relu<<<linBlocks, 160, 0, stream>>>(agg, cnt, h1, W2l, W2r,
                                                      b2, out, nTiles, N);
}